// VQC_47450798686265
// MI455X (gfx1250) — compile-verified
//
#include <hip/hip_runtime.h>

// ---------------------------------------------------------------------------
// 12-qubit VQC, restructured:
//   psi' [s] = prod-state[idx(s)],  idx(s) = s ^ (s>>1) ^ ((s&1)*0xC00)
//   prod-state = U (bits 11..6) (x) V (bits 5..0)   -- Kronecker outer product
//   z_w = cos(th_c)*(n0-n1) + 2 sin(th_c)*(cos(om_c)*ReX + sin(om_c)*ImX)
// One block (128 threads = 4 wave32) per batch element. Outer product done
// with V_WMMA_F32_16X16X4_F32 (16 tiles x {re,im} = 32 WMMAs / block).
// ---------------------------------------------------------------------------

typedef __attribute__((ext_vector_type(2))) float v2f;
typedef __attribute__((ext_vector_type(8))) float v8f;

#define NSTATE 4096
#define THREADS 128

__global__ __launch_bounds__(THREADS)
void vqc_kernel(const float* __restrict__ instance,        // [B,12,3]
                const int*   __restrict__ concept_index,   // [B]
                const float* __restrict__ concept_weights, // [192,12]
                float*       __restrict__ out)             // [B,4]
{
  __shared__ float gR[12][2], gI[12][2];   // first-column gate entries, by BIT p (wire 11-p)
  __shared__ float Ur[64], Ui[64];         // product over bits 6..11 (0.5x prescaled)
  __shared__ float Vr[64], Vi[64];         // product over bits 0..5
  __shared__ float obsA[4], obsB[4], obsC[4];
  __shared__ float psiR[NSTATE];
  __shared__ float psiI[NSTATE];
  __shared__ float partial[THREADS][17];   // padded to dodge bank conflicts
  __shared__ float sums[16];

  const int b   = blockIdx.x;
  const int tid = threadIdx.x;

  // hint the tiny per-block inputs into cache (emits global_prefetch_b8)
  __builtin_prefetch(instance + (size_t)b * 36, 0, 3);

  // ---- Phase 1: gate columns (threads 0-11) and observables (threads 12-15)
  if (tid < 12) {
    const int w = tid;                       // wire w -> bit p = 11 - w
    const float* a = instance + ((size_t)b * 12 + w) * 3;
    const float phi = a[0], th = a[1], om = a[2];
    float c, s;   sincosf(0.5f * th, &s, &c);
    // column0 of Rot(phi,th,om): (e^{-i(phi+om)/2} c ,  e^{-i(phi-om)/2} s)
    float s0, c0; sincosf(-0.5f * (phi + om), &s0, &c0);
    float s1, c1; sincosf( 0.5f * (om  - phi), &s1, &c1);
    const int p = 11 - w;
    gR[p][0] = c * c0;  gI[p][0] = c * s0;
    gR[p][1] = s * c1;  gI[p][1] = s * s1;
  } else if (tid < 16) {
    const int d  = tid - 12;                 // concept slot d -> wire 3d -> bit 11-3d
    const int ci = concept_index[b];
    const float* cw = concept_weights + (size_t)ci * 12 + d * 3;
    const float cth = cw[1], com = cw[2];    // concept phi cancels in <Z>
    float sth, vcth; sincosf(cth, &sth, &vcth);
    float so,  co;   sincosf(com, &so,  &co);
    obsA[d] = vcth;                 // O00 = cos, O11 = -cos
    obsB[d] = 2.0f * sth * co;      // weight of Re(cross)
    obsC[d] = 2.0f * sth * so;      // weight of Im(cross)
  }
  __syncthreads();

  // ---- Phase 2: U[h] (threads 0-63, bits 6..11) and V[l] (threads 64-127, bits 0..5)
  {
    const int h    = tid & 63;
    const int base = (tid < 64) ? 6 : 0;
    float pr = 1.0f, pi = 0.0f;
#pragma unroll
    for (int j = 0; j < 6; ++j) {
      const int t = (h >> j) & 1;
      const float gr = gR[base + j][t], gi = gI[base + j][t];
      const float nr = pr * gr - pi * gi;
      const float ni = pr * gi + pi * gr;
      pr = nr; pi = ni;
    }
    if (tid < 64) { Ur[h] = 0.5f * pr; Ui[h] = 0.5f * pi; }  // 0.5: K lanes duplicated
    else          { Vr[h] = pr;        Vi[h] = pi;        }
  }
  __syncthreads();

  // ---- Phase 3: psi = U (x) V via WMMA rank-2 outer products.
  // Wave v owns row-block I=v; loops column-blocks J=0..3.
  // A/B pairs duplicated across both K halves; 0.5 prescale in U compensates.
  {
    const int wave = tid >> 5;
    const int lane = tid & 31;
    const int m    = lane & 15;
    const int I    = wave;
    const v2f A = { Ur[16 * I + m], Ui[16 * I + m] };
#pragma unroll
    for (int J = 0; J < 4; ++J) {
      const float vr = Vr[16 * J + m];
      const float vi = Vi[16 * J + m];
      const v2f Bre = { vr, -vi };
      const v2f Bim = { vi,  vr };
      const v8f z0 = { 0.f, 0.f, 0.f, 0.f, 0.f, 0.f, 0.f, 0.f };
      v8f dre = __builtin_amdgcn_wmma_f32_16x16x4_f32(false, A, false, Bre,
                                                      (short)0, z0, false, false);
      v8f dim = __builtin_amdgcn_wmma_f32_16x16x4_f32(false, A, false, Bim,
                                                      (short)0, z0, false, false);
      // D layout: reg r / lanes 0-15 -> M=r ; lanes 16-31 -> M=r+8 ; N = lane%16
      const int hbase = 16 * I + ((lane & 16) ? 8 : 0);
      const int lg    = 16 * J + m;
#pragma unroll
      for (int r = 0; r < 8; ++r) {
        psiR[(hbase + r) * 64 + lg] = dre[r];
        psiI[(hbase + r) * 64 + lg] = dim[r];
      }
    }
  }
  __syncthreads();

  // ---- Phase 4: per-wire reductions over all 4096 basis states
  float acc[16];
#pragma unroll
  for (int q = 0; q < 16; ++q) acc[q] = 0.0f;

  const int bits[4] = { 11, 8, 5, 2 };      // CONC_WIRES (0,3,6,9) -> bits
  for (int i = 0; i < NSTATE / THREADS; ++i) {
    const int s   = tid + i * THREADS;
    const int idx = s ^ (s >> 1) ^ ((s & 1) ? 0xC00 : 0);
    const float pr = psiR[idx], pi = psiI[idx];
    const float nrm = pr * pr + pi * pi;
#pragma unroll
    for (int d = 0; d < 4; ++d) {
      const int bb = bits[d];
      const int j  = idx ^ (3 << (bb - 1));  // partner: idx( s ^ (1<<bb) )
      const float qr = psiR[j], qi = psiI[j];
      const bool hi = (s >> bb) & 1;
      if (hi) {
        acc[d * 4 + 1] += nrm;                       // n1
      } else {
        acc[d * 4 + 0] += nrm;                       // n0
        acc[d * 4 + 2] += pr * qr + pi * qi;         // Re conj(p0)*p1
        acc[d * 4 + 3] += pr * qi - pi * qr;         // Im conj(p0)*p1
      }
    }
  }
#pragma unroll
  for (int q = 0; q < 16; ++q) partial[tid][q] = acc[q];
  __syncthreads();

  if (tid < 16) {
    float t = 0.0f;
    for (int i = 0; i < THREADS; ++i) t += partial[i][tid];
    sums[tid] = t;
  }
  __syncthreads();

  if (tid < 4) {
    const int d = tid;
    const float n0  = sums[d * 4 + 0];
    const float n1  = sums[d * 4 + 1];
    const float crR = sums[d * 4 + 2];
    const float crI = sums[d * 4 + 3];
    const float z = obsA[d] * (n0 - n1) + obsB[d] * crR + obsC[d] * crI;
    out[(size_t)b * 4 + d] = 0.5f * z + 0.5f;
  }
}

extern "C" void kernel_launch(void* const* d_in, const int* in_sizes, int n_in,
                              void* d_out, int out_size, void* d_ws, size_t ws_size,
                              hipStream_t stream) {
  const float* instance = (const float*)d_in[0];
  const int*   cidx     = (const int*)d_in[1];
  const float* cweights = (const float*)d_in[2];
  float*       out      = (float*)d_out;
  const int B = in_sizes[0] / 36;            // [B,12,3]
  vqc_kernel<<<B, THREADS, 0, stream>>>(instance, cidx, cweights, out);
}